// CIGINGraphTransformerModel_40587440947796
// MI455X (gfx1250) — compile-verified
//
#include <hip/hip_runtime.h>
#include <math.h>

#define D_     42
#define ED_    10
#define H_     6
#define HD_    7
#define STEPS_ 6
#define B_     12
#define S_     128
#define N_     1536
#define E_     12288
#define PD_    84
#define PH_    14
#define OD_    168
#define ADJW_  48   // N_/32

typedef float v2f __attribute__((ext_vector_type(2)));
typedef float v8f __attribute__((ext_vector_type(8)));

// ---------------------------------------------------------------------------
// Generic f32 GEMM on the WMMA pipe: C[M,Nn] = act(opA(A) @ opB(B) + bias)
// TA: A element (m,k) = A[k*lda+m] else A[m*lda+k]
// TB: B element (k,n) = B[n*ldb+k] else B[k*ldb+n]
// act: 0=none, 1=relu, 2=tanh.  One wave == one 16x16 C tile.
// Out-of-range m/n are handled by index clamping (their C elements are never
// stored); only the K tail needs zero-masking (it pollutes all outputs).
// ---------------------------------------------------------------------------
template <int TA, int TB>
__global__ __launch_bounds__(32)
void gemm_wmma_f32(const float* __restrict__ A, const float* __restrict__ Bm,
                   const float* __restrict__ bias, float* __restrict__ C,
                   int M, int Nn, int K, int lda, int ldb, int ldc, int act)
{
  const int tn = blockIdx.x, tm = blockIdx.y;
  const int lane  = threadIdx.x;
  const int r0    = lane & 15;
  const int khalf = (lane >> 4) * 2;
  const int m_a = tm * 16 + r0;
  const int n_b = tn * 16 + r0;
  const int m_c = (m_a < M) ? m_a : (M - 1);   // clamp: OOB rows never stored
  const int n_c = (n_b < Nn) ? n_b : (Nn - 1); // clamp: OOB cols never stored
  v8f acc = {};
  const int K4 = K & ~3;
  for (int k0 = 0; k0 < K4; k0 += 4) {
    const int ka0 = k0 + khalf, ka1 = ka0 + 1;
    v2f a, b;
    a.x = TA ? A[(size_t)ka0 * lda + m_c] : A[(size_t)m_c * lda + ka0];
    a.y = TA ? A[(size_t)ka1 * lda + m_c] : A[(size_t)m_c * lda + ka1];
    b.x = TB ? Bm[(size_t)n_c * ldb + ka0] : Bm[(size_t)ka0 * ldb + n_c];
    b.y = TB ? Bm[(size_t)n_c * ldb + ka1] : Bm[(size_t)ka1 * ldb + n_c];
    acc = __builtin_amdgcn_wmma_f32_16x16x4_f32(false, a, false, b, (short)0, acc,
                                                false, false);
  }
  if (K4 < K) {  // K tail: clamp index, zero-mask the A fragment
    const int ka0 = K4 + khalf, ka1 = ka0 + 1;
    const int kc0 = (ka0 < K) ? ka0 : (K - 1);
    const int kc1 = (ka1 < K) ? ka1 : (K - 1);
    const float f0 = (ka0 < K) ? 1.f : 0.f;
    const float f1 = (ka1 < K) ? 1.f : 0.f;
    v2f a, b;
    a.x = (TA ? A[(size_t)kc0 * lda + m_c] : A[(size_t)m_c * lda + kc0]) * f0;
    a.y = (TA ? A[(size_t)kc1 * lda + m_c] : A[(size_t)m_c * lda + kc1]) * f1;
    b.x = TB ? Bm[(size_t)n_c * ldb + kc0] : Bm[(size_t)kc0 * ldb + n_c];
    b.y = TB ? Bm[(size_t)n_c * ldb + kc1] : Bm[(size_t)kc1 * ldb + n_c];
    acc = __builtin_amdgcn_wmma_f32_16x16x4_f32(false, a, false, b, (short)0, acc,
                                                false, false);
  }
  const int n  = tn * 16 + (lane & 15);
  const int mb = tm * 16 + ((lane >> 4) ? 8 : 0);
#pragma unroll
  for (int r = 0; r < 8; ++r) {
    const int m = mb + r;
    if (m < M && n < Nn) {
      float v = acc[r];
      if (bias) v += bias[n];
      if (act == 1) v = fmaxf(v, 0.f);
      else if (act == 2) v = tanhf(v);
      C[(size_t)m * ldc + n] = v;
    }
  }
}

// ---------------------------------------------------------------------------
// Per-head masked attention scores via WMMA:
// S[h,i,j] = adj(i,j) ? (Q_h[i]·K_h[j])/sqrt(7) : -1e30
// grid = (N/16, N/16, H), block 32.
// ---------------------------------------------------------------------------
__global__ __launch_bounds__(32)
void attn_scores(const float* __restrict__ Q, const float* __restrict__ Kb,
                 const unsigned* __restrict__ adj, float* __restrict__ S)
{
  const int tn = blockIdx.x, tm = blockIdx.y, h = blockIdx.z;
  const int lane  = threadIdx.x;
  const int r0    = lane & 15;
  const int khalf = (lane >> 4) * 2;
  const int mrow  = tm * 16 + r0;
  const int ncol  = tn * 16 + r0;
  const int qb    = h * HD_;
  v8f acc = {};
#pragma unroll
  for (int k0 = 0; k0 < 8; k0 += 4) {   // HD_=7 padded to 8
    const int ka0 = k0 + khalf, ka1 = ka0 + 1;
    const int kc0 = (ka0 < HD_) ? ka0 : (HD_ - 1);
    const int kc1 = (ka1 < HD_) ? ka1 : (HD_ - 1);
    const float f0 = (ka0 < HD_) ? 1.f : 0.f;
    const float f1 = (ka1 < HD_) ? 1.f : 0.f;
    v2f a, b;
    a.x = Q[mrow * D_ + qb + kc0] * f0;
    a.y = Q[mrow * D_ + qb + kc1] * f1;
    b.x = Kb[ncol * D_ + qb + kc0];
    b.y = Kb[ncol * D_ + qb + kc1];
    acc = __builtin_amdgcn_wmma_f32_16x16x4_f32(false, a, false, b, (short)0, acc,
                                                false, false);
  }
  const float scale = 0.3779644730092272f; // 1/sqrt(7)
  const int n  = tn * 16 + (lane & 15);
  const int mb = tm * 16 + ((lane >> 4) ? 8 : 0);
  const size_t base = (size_t)h * N_ * N_;
#pragma unroll
  for (int r = 0; r < 8; ++r) {
    const int m = mb + r;
    const unsigned w = adj[m * ADJW_ + (n >> 5)];
    const float sv = ((w >> (n & 31)) & 1u) ? acc[r] * scale : -1e30f;
    S[base + (size_t)m * N_ + n] = sv;
  }
}

// ---------------------------------------------------------------------------
// out[i, h*7+d] = sum_j attn[h,i,j] * V[j, h*7+d]   (K = N = 1536 WMMA loop)
// Two independent accumulator chains to hide XDL latency.
// grid = (1, N/16, H), block 32.
// ---------------------------------------------------------------------------
__global__ __launch_bounds__(32)
void attn_av(const float* __restrict__ S, const float* __restrict__ V,
             float* __restrict__ out)
{
  const int tm = blockIdx.y, h = blockIdx.z;
  const int lane  = threadIdx.x;
  const int r0    = lane & 15;
  const int khalf = (lane >> 4) * 2;
  const int mrow  = tm * 16 + r0;
  const int vcol  = h * HD_ + ((r0 < HD_) ? r0 : (HD_ - 1)); // clamp: cols 7..15 unused
  const float* __restrict__ Srow = S + (size_t)h * N_ * N_ + (size_t)mrow * N_;
  v8f acc0 = {}, acc1 = {};
  for (int k0 = 0; k0 < N_; k0 += 8) {
    const int ka = k0 + khalf;
    const int kb = ka + 4;
    v2f a0, b0, a1, b1;
    a0.x = Srow[ka];     a0.y = Srow[ka + 1];
    b0.x = V[ka * D_ + vcol];
    b0.y = V[(ka + 1) * D_ + vcol];
    a1.x = Srow[kb];     a1.y = Srow[kb + 1];
    b1.x = V[kb * D_ + vcol];
    b1.y = V[(kb + 1) * D_ + vcol];
    acc0 = __builtin_amdgcn_wmma_f32_16x16x4_f32(false, a0, false, b0, (short)0, acc0,
                                                 false, false);
    acc1 = __builtin_amdgcn_wmma_f32_16x16x4_f32(false, a1, false, b1, (short)0, acc1,
                                                 false, false);
  }
  acc0 = acc0 + acc1;
  const int n  = lane & 15;
  const int mb = tm * 16 + ((lane >> 4) ? 8 : 0);
  if (n < HD_) {
#pragma unroll
    for (int r = 0; r < 8; ++r)
      out[(mb + r) * D_ + h * HD_ + n] = acc0[r];
  }
}

// --- edge-bias scatter (set semantics: recompute QK dot, add eb) -----------
__global__ void scatter_eb(const float* __restrict__ Q, const float* __restrict__ Kb,
                           const float* __restrict__ eb, const int* __restrict__ src,
                           const int* __restrict__ dst, float* __restrict__ S)
{
  const int e = blockIdx.x * blockDim.x + threadIdx.x;
  if (e >= E_) return;
  const int i = src[e], j = dst[e];
  const float scale = 0.3779644730092272f;
  for (int h = 0; h < H_; ++h) {
    float dot = 0.f;
    for (int d = 0; d < HD_; ++d)
      dot += Q[i * D_ + h * HD_ + d] * Kb[j * D_ + h * HD_ + d];
    S[(size_t)h * N_ * N_ + (size_t)i * N_ + j] = dot * scale + eb[e * H_ + h];
  }
}

// --- adjacency bitmask ------------------------------------------------------
__global__ void zero_u32(unsigned* p, int n)
{
  const int t = blockIdx.x * blockDim.x + threadIdx.x;
  if (t < n) p[t] = 0u;
}
__global__ void adj_edges(unsigned* adj, const int* __restrict__ src,
                          const int* __restrict__ dst, int E)
{
  const int e = blockIdx.x * blockDim.x + threadIdx.x;
  if (e >= E) return;
  const int s = src[e], d = dst[e];
  atomicOr(&adj[s * ADJW_ + (d >> 5)], 1u << (d & 31));
  atomicOr(&adj[d * ADJW_ + (s >> 5)], 1u << (s & 31));
}
__global__ void adj_diag(unsigned* adj)
{
  const int i = blockIdx.x * blockDim.x + threadIdx.x;
  if (i < N_) adj[i * ADJW_ + (i >> 5)] |= 1u << (i & 31);
}

// --- row softmax over j for S[h,i,:] ---------------------------------------
__global__ __launch_bounds__(256)
void softmax_rows(float* __restrict__ S)
{
  const int i = blockIdx.x, h = blockIdx.y;
  float* row = S + ((size_t)h * N_ + i) * (size_t)N_;
  __shared__ float red[256];
  const int t = threadIdx.x;
  float m = -1e30f;
  for (int j = t; j < N_; j += 256) m = fmaxf(m, row[j]);
  red[t] = m; __syncthreads();
  for (int s = 128; s > 0; s >>= 1) { if (t < s) red[t] = fmaxf(red[t], red[t + s]); __syncthreads(); }
  const float mx = red[0]; __syncthreads();
  float sum = 0.f;
  for (int j = t; j < N_; j += 256) { float e = __expf(row[j] - mx); row[j] = e; sum += e; }
  red[t] = sum; __syncthreads();
  for (int s = 128; s > 0; s >>= 1) { if (t < s) red[t] += red[t + s]; __syncthreads(); }
  const float inv = 1.f / red[0];
  for (int j = t; j < N_; j += 256) row[j] *= inv;
}

// --- fused residual-add + LayerNorm over D_=42 -----------------------------
__global__ __launch_bounds__(64)
void add_ln(const float* __restrict__ a, const float* __restrict__ b,
            const float* __restrict__ g, const float* __restrict__ be,
            float* __restrict__ out)
{
  const int i = blockIdx.x, t = threadIdx.x;
  __shared__ float red[64];
  float v = 0.f;
  if (t < D_) v = a[i * D_ + t] + b[i * D_ + t];
  red[t] = v; __syncthreads();
  for (int s = 32; s > 0; s >>= 1) { if (t < s) red[t] += red[t + s]; __syncthreads(); }
  const float mean = red[0] / (float)D_; __syncthreads();
  const float c = (t < D_) ? (v - mean) : 0.f;
  red[t] = c * c; __syncthreads();
  for (int s = 32; s > 0; s >>= 1) { if (t < s) red[t] += red[t + s]; __syncthreads(); }
  const float inv = rsqrtf(red[0] / (float)D_ + 1e-5f);
  if (t < D_) out[i * D_ + t] = c * inv * g[t] + be[t];
}

// --- small helpers ----------------------------------------------------------
__global__ void copy_mat(const float* __restrict__ src, int ls,
                         float* __restrict__ dst, int ld, int rows, int cols)
{
  const int tid = blockIdx.x * blockDim.x + threadIdx.x;
  if (tid >= rows * cols) return;
  const int r = tid / cols, c = tid % cols;
  dst[(size_t)r * ld + c] = src[(size_t)r * ls + c];
}
__global__ void add_vec(const float* __restrict__ a, const float* __restrict__ b,
                        float* __restrict__ c, int n)
{
  const int t = blockIdx.x * blockDim.x + threadIdx.x;
  if (t < n) c[t] = a[t] + b[t];
}

// --- pooling: 1-query MHA per graph segment --------------------------------
__global__ void pool_q(const float* __restrict__ query, const float* __restrict__ Wq,
                       const float* __restrict__ bq, float* __restrict__ q1)
{
  const int t = threadIdx.x;
  if (t >= PD_) return;
  float s = bq[t];
  for (int d = 0; d < PD_; ++d) s += query[d] * Wq[t * PD_ + d];
  q1[t] = s;
}
__global__ void pool_sc(const float* __restrict__ q1, const float* __restrict__ pk,
                        float* __restrict__ sc)
{
  const int tid = blockIdx.x * blockDim.x + threadIdx.x;
  if (tid >= B_ * H_ * S_) return;
  const int b = tid / (H_ * S_), rem = tid % (H_ * S_);
  const int h = rem / S_, s = rem % S_;
  float dot = 0.f;
  for (int d = 0; d < PH_; ++d)
    dot += q1[h * PH_ + d] * pk[(b * S_ + s) * PD_ + h * PH_ + d];
  sc[(b * H_ + h) * S_ + s] = dot * 0.2672612419124244f; // 1/sqrt(14)
}
__global__ __launch_bounds__(128)
void pool_softmax(float* __restrict__ sc)
{
  float* row = sc + blockIdx.x * S_;
  __shared__ float red[128];
  const int t = threadIdx.x;
  const float v = row[t];
  red[t] = v; __syncthreads();
  for (int s = 64; s > 0; s >>= 1) { if (t < s) red[t] = fmaxf(red[t], red[t + s]); __syncthreads(); }
  const float mx = red[0]; __syncthreads();
  const float e = __expf(v - mx);
  red[t] = e; __syncthreads();
  for (int s = 64; s > 0; s >>= 1) { if (t < s) red[t] += red[t + s]; __syncthreads(); }
  row[t] = e / red[0];
}
__global__ void pool_av(const float* __restrict__ sc, const float* __restrict__ pv,
                        float* __restrict__ po)
{
  const int tid = blockIdx.x * blockDim.x + threadIdx.x;
  if (tid >= B_ * PD_) return;
  const int b = tid / PD_, c = tid % PD_, h = c / PH_;
  float s = 0.f;
  for (int j = 0; j < S_; ++j)
    s += sc[(b * H_ + h) * S_ + j] * pv[(b * S_ + j) * PD_ + c];
  po[b * PD_ + c] = s;
}

// ===========================================================================
struct TowerP {
  const float *Wi, *bi, *Wq, *bq, *Wk, *bk, *Wv, *bv, *Wo, *bo, *We, *be;
  const float *g1, *lb1, *g2, *lb2, *W1, *b1, *W2, *b2;
};
struct PoolP {
  const float *Wq, *bq, *Wk, *bk, *Wv, *bv, *Wo, *bo, *query, *Wp, *bp;
};

extern "C" void kernel_launch(void* const* d_in, const int* in_sizes, int n_in,
                              void* d_out, int out_size, void* d_ws, size_t ws_size,
                              hipStream_t stream)
{
  (void)in_sizes; (void)n_in; (void)out_size; (void)ws_size;
  auto F = [&](int i) { return (const float*)d_in[i]; };

  const float* x_u  = F(0);
  const float* ef_u = F(1);
  const float* x_v  = F(2);
  const float* ef_v = F(3);
  TowerP PU = { F(4), F(5), F(6), F(7), F(8), F(9), F(10), F(11), F(12), F(13),
                F(14), F(15), F(16), F(17), F(18), F(19), F(20), F(21), F(22), F(23) };
  TowerP PV = { F(24), F(25), F(26), F(27), F(28), F(29), F(30), F(31), F(32), F(33),
                F(34), F(35), F(36), F(37), F(38), F(39), F(40), F(41), F(42), F(43) };
  PoolP PLU = { F(44), F(45), F(46), F(47), F(48), F(49), F(50), F(51), F(52), F(53), F(54) };
  PoolP PLV = { F(55), F(56), F(57), F(58), F(59), F(60), F(61), F(62), F(63), F(64), F(65) };
  const float* fcW1 = F(66); const float* fcb1 = F(67);
  const float* fcW2 = F(68); const float* fcb2 = F(69);
  const float* fcW3 = F(70); const float* fcb3 = F(71);
  const int* src_u = (const int*)d_in[72];
  const int* dst_u = (const int*)d_in[73];
  const int* src_v = (const int*)d_in[74];
  const int* dst_v = (const int*)d_in[75];

  // ---- workspace bump allocator (~62 MB total) ----
  float* wsf = (float*)d_ws;
  size_t off = 0;
  auto alloc = [&](size_t nf) -> float* {
    float* p = wsf + off; off += (nf + 63) & ~(size_t)63; return p;
  };
  unsigned* adj = (unsigned*)alloc(N_ * ADJW_);
  float* Qb   = alloc((size_t)N_ * D_);
  float* Kb   = alloc((size_t)N_ * D_);
  float* Vb   = alloc((size_t)N_ * D_);
  float* ebuf = alloc((size_t)E_ * H_);
  float* Sb   = alloc((size_t)H_ * N_ * N_);
  float* t1   = alloc((size_t)N_ * D_);
  float* t2   = alloc((size_t)N_ * D_);
  float* ffn  = alloc((size_t)N_ * 4 * D_);
  float* hA   = alloc((size_t)N_ * D_);
  float* hB   = alloc((size_t)N_ * D_);
  float* hini = alloc((size_t)N_ * D_);
  float* hu   = alloc((size_t)N_ * D_);
  float* hv   = alloc((size_t)N_ * D_);
  float* hcu  = alloc((size_t)N_ * 2 * D_);
  float* hcv  = alloc((size_t)N_ * 2 * D_);
  float* pk   = alloc((size_t)N_ * PD_);
  float* pvv  = alloc((size_t)N_ * PD_);
  float* q1   = alloc(PD_);
  float* scb  = alloc(B_ * H_ * S_);
  float* po   = alloc(B_ * PD_);
  float* po2  = alloc(B_ * PD_);
  float* fcat = alloc(B_ * 8 * D_);
  float* fh1  = alloc(B_ * 256);
  float* fh2  = alloc(B_ * 128);

  auto gemm = [&](const float* A, const float* Bm, const float* bias, float* C,
                  int M, int Nn, int K, int lda, int ldb, int ldc,
                  int tA, int tB, int act) {
    dim3 g((Nn + 15) / 16, (M + 15) / 16, 1);
    if (!tA && !tB)
      gemm_wmma_f32<0, 0><<<g, 32, 0, stream>>>(A, Bm, bias, C, M, Nn, K, lda, ldb, ldc, act);
    else if (!tA && tB)
      gemm_wmma_f32<0, 1><<<g, 32, 0, stream>>>(A, Bm, bias, C, M, Nn, K, lda, ldb, ldc, act);
    else if (tA && !tB)
      gemm_wmma_f32<1, 0><<<g, 32, 0, stream>>>(A, Bm, bias, C, M, Nn, K, lda, ldb, ldc, act);
    else
      gemm_wmma_f32<1, 1><<<g, 32, 0, stream>>>(A, Bm, bias, C, M, Nn, K, lda, ldb, ldc, act);
  };

  auto gather = [&](const float* x, const float* ef, const int* src, const int* dst,
                    const TowerP& P, float* hout) {
    // h = relu(x @ Wi + bi); init = h
    gemm(x, P.Wi, P.bi, hA, N_, D_, D_, D_, D_, D_, 0, 0, 1);
    copy_mat<<<(N_ * D_ + 255) / 256, 256, 0, stream>>>(hA, D_, hini, D_, N_, D_);
    // adjacency bitmask
    zero_u32<<<(N_ * ADJW_ + 255) / 256, 256, 0, stream>>>(adj, N_ * ADJW_);
    adj_edges<<<(E_ + 255) / 256, 256, 0, stream>>>(adj, src, dst, E_);
    adj_diag<<<(N_ + 255) / 256, 256, 0, stream>>>(adj);
    for (int l = 0; l < STEPS_; ++l) {
      gemm(hA, P.Wq + l * D_ * D_, P.bq + l * D_, Qb, N_, D_, D_, D_, D_, D_, 0, 0, 0);
      gemm(hA, P.Wk + l * D_ * D_, P.bk + l * D_, Kb, N_, D_, D_, D_, D_, D_, 0, 0, 0);
      gemm(hA, P.Wv + l * D_ * D_, P.bv + l * D_, Vb, N_, D_, D_, D_, D_, D_, 0, 0, 0);
      gemm(ef, P.We + l * ED_ * H_, P.be + l * H_, ebuf, E_, H_, ED_, ED_, H_, H_, 0, 0, 0);
      attn_scores<<<dim3(N_ / 16, N_ / 16, H_), 32, 0, stream>>>(Qb, Kb, adj, Sb);
      scatter_eb<<<(E_ + 255) / 256, 256, 0, stream>>>(Qb, Kb, ebuf, src, dst, Sb);
      softmax_rows<<<dim3(N_, H_), 256, 0, stream>>>(Sb);
      attn_av<<<dim3(1, N_ / 16, H_), 32, 0, stream>>>(Sb, Vb, t1);
      gemm(t1, P.Wo + l * D_ * D_, P.bo + l * D_, t2, N_, D_, D_, D_, D_, D_, 0, 0, 0);
      add_ln<<<N_, 64, 0, stream>>>(t2, hA, P.g1 + l * D_, P.lb1 + l * D_, hB);
      gemm(hB, P.W1 + l * D_ * 4 * D_, P.b1 + l * 4 * D_, ffn,
           N_, 4 * D_, D_, D_, 4 * D_, 4 * D_, 0, 0, 1);
      gemm(ffn, P.W2 + l * 4 * D_ * D_, P.b2 + l * D_, t1,
           N_, D_, 4 * D_, 4 * D_, D_, D_, 0, 0, 0);
      add_ln<<<N_, 64, 0, stream>>>(hB, t1, P.g2 + l * D_, P.lb2 + l * D_, hA);
    }
    add_vec<<<(N_ * D_ + 255) / 256, 256, 0, stream>>>(hA, hini, hout, N_ * D_);
  };

  auto pool = [&](const float* hcat, const PoolP& P, float* fdst) {
    gemm(hcat, P.Wk, P.bk, pk, N_, PD_, PD_, PD_, PD_, PD_, 0, 1, 0);
    gemm(hcat, P.Wv, P.bv, pvv, N_, PD_, PD_, PD_, PD_, PD_, 0, 1, 0);
    pool_q<<<1, 128, 0, stream>>>(P.query, P.Wq, P.bq, q1);
    pool_sc<<<(B_ * H_ * S_ + 255) / 256, 256, 0, stream>>>(q1, pk, scb);
    pool_softmax<<<B_ * H_, 128, 0, stream>>>(scb);
    pool_av<<<(B_ * PD_ + 255) / 256, 256, 0, stream>>>(scb, pvv, po);
    gemm(po, P.Wo, P.bo, po2, B_, PD_, PD_, PD_, PD_, PD_, 0, 1, 0);
    gemm(po2, P.Wp, P.bp, fdst, B_, OD_, PD_, PD_, OD_, 2 * OD_, 0, 0, 0);
  };

  // ---- towers ----
  gather(x_u, ef_u, src_u, dst_u, PU, hu);
  gather(x_v, ef_v, src_v, dst_v, PV, hv);

  // ---- interaction: inter = tanh(hu @ hv^T) written straight to output ----
  float* inter = (float*)d_out + B_;   // outputs: pred[12] then inter[N*N]
  gemm(hu, hv, nullptr, inter, N_, N_, D_, D_, D_, N_, 0, 1, 2);
  // hcat_u = [hu, inter @ hv]; hcat_v = [hv, inter^T @ hu]
  copy_mat<<<(N_ * D_ + 255) / 256, 256, 0, stream>>>(hu, D_, hcu, 2 * D_, N_, D_);
  gemm(inter, hv, nullptr, hcu + D_, N_, D_, N_, N_, D_, 2 * D_, 0, 0, 0);
  copy_mat<<<(N_ * D_ + 255) / 256, 256, 0, stream>>>(hv, D_, hcv, 2 * D_, N_, D_);
  gemm(inter, hu, nullptr, hcv + D_, N_, D_, N_, N_, D_, 2 * D_, 1, 0, 0);

  // ---- pooling (writes both halves of fcat = [pu, pv]) ----
  pool(hcu, PLU, fcat);
  pool(hcv, PLV, fcat + OD_);

  // ---- MLP head -> pred into d_out[0..11] ----
  gemm(fcat, fcW1, fcb1, fh1, B_, 256, 8 * D_, 8 * D_, 256, 256, 0, 0, 1);
  gemm(fh1, fcW2, fcb2, fh2, B_, 128, 256, 256, 128, 128, 0, 0, 1);
  gemm(fh2, fcW3, fcb3, (float*)d_out, B_, 1, 128, 128, 1, 1, 0, 0, 0);
}